// GraphLaplacianLoss_1211180777906
// MI455X (gfx1250) — compile-verified
//
#include <hip/hip_runtime.h>

// Problem constants (match reference)
#define NV 100000
#define NE 3200000
#define NB 8
#define NC 3
#define ROW 24            // NB*NC contiguous floats per vertex
#define NELEM (NV * ROW)  // 2,400,000 (divisible by 256)
#define RED_BLOCKS 512

typedef __attribute__((ext_vector_type(2))) float v2f;
typedef __attribute__((ext_vector_type(8))) float v8f;

// ---- fp32 hardware atomic add (global_atomic_add_f32, no CAS loop) ----
__device__ __forceinline__ void atomAddF(float* p, float v) {
    unsafeAtomicAdd(p, v);
}

// ---- Wave32 reduction via V_WMMA_F32_16X16X4_F32 ----
// A: lane m (<16) -> A[m][0]=x, lane m+16 -> A[m][2]=x, K=1,3 zero.
// B: all ones (every lane, both VGPRs) -> every B[k][n]=1.
// D[m][n] = sum_k A[m][k] = x[m] + x[m+16]  (independent of n).
// C/D layout: lanes 0-15 hold rows M=0..7 in VGPRs 0..7, lanes 16-31 rows M=8..15.
// Sum the 8 D regs per lane, then xor-16 combine -> full 32-lane sum in all lanes.
__device__ __forceinline__ float wave_reduce_wmma(float x) {
    v2f a; a.x = x;    a.y = 0.0f;
    v2f b; b.x = 1.0f; b.y = 1.0f;
    v8f c = {};
    v8f d = __builtin_amdgcn_wmma_f32_16x16x4_f32(
        /*neg_a=*/false, a, /*neg_b=*/false, b,
        /*c_mod=*/(short)0, c, /*reuse_a=*/false, /*reuse_b=*/false);
    float s = ((d[0] + d[1]) + (d[2] + d[3])) + ((d[4] + d[5]) + (d[6] + d[7]));
    s += __shfl_xor(s, 16, 32);
    return s;
}

__device__ __forceinline__ float block_reduce(float s) {
    __shared__ float wsum[8];
    s = wave_reduce_wmma(s);
    const int lane = threadIdx.x & 31, wid = threadIdx.x >> 5;
    if (lane == 0) wsum[wid] = s;
    __syncthreads();
    float t = 0.0f;
    if (threadIdx.x == 0) {
#pragma unroll
        for (int k = 0; k < 8; ++k) t += wsum[k];
    }
    return t;  // valid in thread 0 only
}

// ---- 1) zero degree histogram ----
__global__ __launch_bounds__(256) void k_zero_deg(unsigned* __restrict__ deg) {
    int i = blockIdx.x * 256 + threadIdx.x;
    if (i < NV) deg[i] = 0u;
}

// ---- 2) degree = segment_sum(ones, row) : integer atomics (exact) ----
__global__ __launch_bounds__(256) void k_count_deg(const int* __restrict__ ei,
                                                   unsigned* __restrict__ deg) {
    int e = blockIdx.x * 256 + threadIdx.x;  // NE divisible by 256
    atomicAdd(&deg[ei[e]], 1u);
}

// ---- 3) dinv = deg>0 ? deg^-1/2 : 0 ----
__global__ __launch_bounds__(256) void k_dinv(const unsigned* __restrict__ deg,
                                              float* __restrict__ dinv) {
    int i = blockIdx.x * 256 + threadIdx.x;
    if (i < NV) {
        unsigned dgi = deg[i];
        dinv[i] = (dgi > 0u) ? rsqrtf((float)dgi) : 0.0f;
    }
}

// ---- 4) diff[v][b*3+c] = x - y  (transpose (B,V,C) -> (V,24)); acc init = self-loop ----
__global__ __launch_bounds__(256) void k_diff(const float* __restrict__ f,
                                              const float* __restrict__ t,
                                              float* __restrict__ diff,
                                              float* __restrict__ acc) {
    int i = blockIdx.x * 256 + threadIdx.x;  // NELEM divisible by 256
    int v = i / ROW, j = i % ROW;
    int b = j / NC, c = j % NC;
    size_t src = ((size_t)b * NV + (size_t)v) * NC + (size_t)c;
    float d = f[src] - t[src];
    diff[i] = d;
    acc[i]  = d;  // self-loop weight = 1
}

// ---- 5) edge scatter: acc[col] += (-dinv[row]*dinv[col]) * diff[row]  (24 floats) ----
__global__ __launch_bounds__(256) void k_scatter(const int* __restrict__ ei,
                                                 const float* __restrict__ dinv,
                                                 const float* __restrict__ diff,
                                                 float* __restrict__ acc) {
    int e = blockIdx.x * 256 + threadIdx.x;  // NE divisible by 256
    int r = ei[e];
    int c = ei[NE + e];
    float w = -dinv[r] * dinv[c];
    const float4* __restrict__ src = (const float4*)(diff + (size_t)r * ROW);
    float* __restrict__ dst = acc + (size_t)c * ROW;
#pragma unroll
    for (int q = 0; q < 6; ++q) {
        float4 v = src[q];
        atomAddF(dst + q * 4 + 0, w * v.x);
        atomAddF(dst + q * 4 + 1, w * v.y);
        atomAddF(dst + q * 4 + 2, w * v.z);
        atomAddF(dst + q * 4 + 3, w * v.w);
    }
}

// ---- 6) per-block sum of squares (WMMA wave reduction) ----
__global__ __launch_bounds__(256) void k_reduce(const float* __restrict__ acc,
                                                float* __restrict__ partials) {
    float s = 0.0f;
    for (int i = blockIdx.x * 256 + threadIdx.x; i < NELEM; i += RED_BLOCKS * 256) {
        float v = acc[i];
        s = fmaf(v, v, s);
    }
    float t = block_reduce(s);
    if (threadIdx.x == 0) partials[blockIdx.x] = t;
}

// ---- 7) finalize: mean ----
__global__ __launch_bounds__(256) void k_final(const float* __restrict__ partials,
                                               float* __restrict__ out) {
    float s = 0.0f;
    for (int i = threadIdx.x; i < RED_BLOCKS; i += 256) s += partials[i];
    float t = block_reduce(s);
    if (threadIdx.x == 0) out[0] = t * (1.0f / (float)NELEM);
}

extern "C" void kernel_launch(void* const* d_in, const int* in_sizes, int n_in,
                              void* d_out, int out_size, void* d_ws, size_t ws_size,
                              hipStream_t stream) {
    const float* f  = (const float*)d_in[0];  // (B,V,C) float32
    const float* t  = (const float*)d_in[1];  // (B,V,C) float32
    const int*   ei = (const int*)d_in[2];    // (2,E)   int32

    // Workspace layout (bytes):
    //   deg:      [0,           400,000)    uint32[V]
    //   dinv:     [400,000,     800,000)    float[V]
    //   diff:     [800,000,     10,400,000) float[V*24]   (16B aligned: 800000%16==0)
    //   acc:      [10,400,000,  20,000,000) float[V*24]   (16B aligned)
    //   partials: [20,000,000,  20,002,048) float[512]
    char* ws = (char*)d_ws;
    unsigned* deg      = (unsigned*)(ws);
    float*    dinv     = (float*)(ws + 400000u);
    float*    diff     = (float*)(ws + 800000u);
    float*    acc      = (float*)(ws + 10400000u);
    float*    partials = (float*)(ws + 20000000u);
    float*    out      = (float*)d_out;

    const int vBlocks = (NV + 255) / 256;        // 391
    const int eBlocks = NE / 256;                // 12500
    const int nBlocks = NELEM / 256;             // 9375

    k_zero_deg<<<vBlocks, 256, 0, stream>>>(deg);
    k_count_deg<<<eBlocks, 256, 0, stream>>>(ei, deg);
    k_dinv<<<vBlocks, 256, 0, stream>>>(deg, dinv);
    k_diff<<<nBlocks, 256, 0, stream>>>(f, t, diff, acc);
    k_scatter<<<eBlocks, 256, 0, stream>>>(ei, dinv, diff, acc);
    k_reduce<<<RED_BLOCKS, 256, 0, stream>>>(acc, partials);
    k_final<<<1, 256, 0, stream>>>(partials, out);
}